// LocusEncoder_4690104287245
// MI455X (gfx1250) — compile-verified
//
#include <hip/hip_runtime.h>
#include <hip/hip_bf16.h>

// ---------------------------------------------------------------------------
// LocusEncoder: R = [W_L @ X_L ; W_R @ X_R];  out = pairedcorr(R, X)
// Uncentered GEMMs + rank-1 mean correction:
//   num[i][j]   = (A @ X^T)[i][j] - T * meanA[i] * meanX[j]
//   normAc[i]^2 = sumsq(A[i]) - T * meanA[i]^2   (same for X rows)
// f32 WMMA (V_WMMA_F32_16X16X4_F32): ~36 GFLOP vs ~260 MB -> L2/HBM bound,
// so full-precision f32 matrix cores are the right path on MI455X.
// Interior tiles staged with gfx1250 async global->LDS (ASYNCcnt) b128 copies.
// ---------------------------------------------------------------------------

typedef __attribute__((ext_vector_type(2))) float v2f;
typedef __attribute__((ext_vector_type(8))) float v8f;

#define N_LOCI 59412
#define TDIM   500
#define NENC   200
#define MTOT   400
#define KL     29696
#define KR     29716
#define SPLITS 8
#define KCHUNK 3776          // ceil(29716/8) rounded up to 64
#define PAD_A  68            // 68*4 = 272 B row stride: 16B aligned, no conflicts
#define PAD_B1 132           // GEMM1 Bs[k][n] rows of 128: 528 B stride, 16B aligned
#define PAD_B2 68

#define USE_ASYNC_LDS 1

#if USE_ASYNC_LDS
// 16B per lane, global -> LDS, tracked by ASYNCcnt. LDS operand is the byte
// offset within the wave's LDS allocation (= low 32 bits of the generic addr).
__device__ __forceinline__ void ld16_to_lds(float* l, const float* g) {
  asm volatile("global_load_async_to_lds_b128 %0, %1, off"
               :: "v"((unsigned int)(unsigned long long)l), "v"(g) : "memory");
}
__device__ __forceinline__ void lds_copy_fence() {
  asm volatile("s_wait_asynccnt 0x0" ::: "memory");
}
#else
__device__ __forceinline__ void ld16_to_lds(float* l, const float* g) {
  *(float4*)l = *(const float4*)g;
}
__device__ __forceinline__ void lds_copy_fence() {}
#endif

__device__ __forceinline__ float wave_sum(float v) {
#pragma unroll
  for (int m = 16; m >= 1; m >>= 1) v += __shfl_xor(v, m, 32);
  return v;
}

// ---- per-row mean / centered norm (one wave32 per row) ---------------------
__global__ void row_stats_kernel(const float* __restrict__ M, int rows,
                                 float* __restrict__ mean,
                                 float* __restrict__ nrm) {
  const int wave = threadIdx.x >> 5;
  const int lane = threadIdx.x & 31;
  const int row  = blockIdx.x * 8 + wave;
  if (row >= rows) return;
  const float* p = M + (size_t)row * TDIM;
  float s = 0.f, ss = 0.f;
  for (int t = lane; t < TDIM; t += 32) {
    float v = p[t];
    s += v;
    ss += v * v;
  }
  s  = wave_sum(s);
  ss = wave_sum(ss);
  if (lane == 0) {
    mean[row] = s * (1.0f / TDIM);
    float c = ss - s * s * (1.0f / TDIM);
    nrm[row] = sqrtf(fmaxf(c, 0.0f));
  }
}

// ---- GEMM1: partial[s] = W_comp @ X_slice over K-chunk s -------------------
// 64(M) x 128(N) block tile, 16 wave32; each wave owns a 16x32 strip
// (two 16x16 WMMA subtiles sharing the A fragment).
__global__ void __launch_bounds__(512)
gemm1_partial_kernel(const float* __restrict__ X,
                     const float* __restrict__ WL,
                     const float* __restrict__ WR,
                     float* __restrict__ P) {
  __shared__ __align__(16) float As[64][PAD_A];    // W tile:  As[m][k]
  __shared__ __align__(16) float Bs[64][PAD_B1];   // X tile:  Bs[k][n]

  const int z    = blockIdx.z;
  const int comp = z / SPLITS;          // 0 = cortex_L, 1 = cortex_R
  const int sp   = z - comp * SPLITS;
  const int Kc   = comp ? KR : KL;
  const int xoff = comp ? KL : 0;
  const float* W = comp ? WR : WL;

  const int row0 = blockIdx.y * 64;     // row inside compartment (pad to 256)
  const int col0 = blockIdx.x * 128;    // time column (pad 500 -> 512)
  const int kbeg = sp * KCHUNK;
  const int kend = min(kbeg + KCHUNK, Kc);

  const int tid  = threadIdx.x;
  const int wv   = tid >> 5;
  const int lane = tid & 31;
  const int half = lane >> 4;
  const int ml   = lane & 15;
  const int mi   = wv >> 2;             // subtile row 0..3
  const int ni0  = (wv & 3) * 2;        // subtile cols {ni0, ni0+1} of 0..7

  const bool rows_in = (row0 + 64 <= NENC);
  const bool cols_in = (col0 + 128 <= TDIM);

  v8f acc0 = {}, acc1 = {};

  for (int kc = kbeg; kc < kend; kc += 64) {
    const bool k_in = (kc + 64 <= kend);
    // ---- stage A: As[m][kk] = W[row0+m][kc+kk] (64 rows x 64 k) ----
    if (rows_in && k_in) {
      const int m  = tid >> 3;
      const int c0 = (tid & 7) * 8;
      const float* g = W + (size_t)(row0 + m) * Kc + kc + c0;
      ld16_to_lds(&As[m][c0], g);
      ld16_to_lds(&As[m][c0 + 4], g + 4);
    } else {
      const int m    = tid >> 3;
      const int c0   = (tid & 7) * 8;
      const int grow = row0 + m;
      const float* wr = W + (size_t)grow * Kc;
#pragma unroll
      for (int i = 0; i < 8; ++i) {
        int k = kc + c0 + i;
        As[m][c0 + i] = (grow < NENC && k < kend) ? wr[k] : 0.0f;
      }
    }
    // ---- stage B: Bs[kk][n] = X[xoff+kc+kk][col0+n] (64 k x 128 n) ----
    if (cols_in && k_in) {
      const int kk = tid >> 3;
      const int n0 = (tid & 7) * 16;
      const float* g = X + (size_t)(xoff + kc + kk) * TDIM + col0 + n0;
#pragma unroll
      for (int i = 0; i < 4; ++i)
        ld16_to_lds(&Bs[kk][n0 + 4 * i], g + 4 * i);
    } else {
      const int kk = tid >> 3;
      const int n0 = (tid & 7) * 16;
      const int k  = kc + kk;
      const float* xr = X + (size_t)(xoff + k) * TDIM;
#pragma unroll
      for (int i = 0; i < 16; ++i) {
        int c = col0 + n0 + i;
        Bs[kk][n0 + i] = (k < kend && c < TDIM) ? xr[c] : 0.0f;
      }
    }
    lds_copy_fence();
    __syncthreads();

    const int na = ni0 * 16 + ml;
    const int nb = na + 16;
#pragma unroll
    for (int k4 = 0; k4 < 64; k4 += 4) {
      const int kk = k4 + 2 * half;
      v2f a = *(const v2f*)&As[mi * 16 + ml][kk];  // A[m][2h], A[m][2h+1]
      v2f b0, b1;
      b0.x = Bs[kk][na];
      b0.y = Bs[kk + 1][na];
      b1.x = Bs[kk][nb];
      b1.y = Bs[kk + 1][nb];
      acc0 = __builtin_amdgcn_wmma_f32_16x16x4_f32(
          false, a, false, b0, (short)0, acc0, false, false);
      acc1 = __builtin_amdgcn_wmma_f32_16x16x4_f32(
          false, a, false, b1, (short)0, acc1, false, false);
    }
    __syncthreads();
  }

  float* Ps = P + (size_t)sp * (MTOT * TDIM);
#pragma unroll
  for (int v = 0; v < 8; ++v) {
    const int rloc = row0 + mi * 16 + v + 8 * half;  // C/D layout: M=v+8*half
    if (rloc < NENC) {
      const size_t rb = (size_t)(comp * NENC + rloc) * TDIM;
      const int c0 = col0 + ni0 * 16 + ml;
      const int c1 = c0 + 16;
      if (c0 < TDIM) Ps[rb + c0] = acc0[v];
      if (c1 < TDIM) Ps[rb + c1] = acc1[v];
    }
  }
}

// ---- fixed-order K-split reduce: R = sum_s P[s] ----------------------------
__global__ void reduce_partials_kernel(const float* __restrict__ P,
                                       float* __restrict__ R) {
  const int i = blockIdx.x * blockDim.x + threadIdx.x;
  if (i >= MTOT * TDIM) return;
  float s = 0.f;
#pragma unroll
  for (int sp = 0; sp < SPLITS; ++sp) s += P[(size_t)sp * (MTOT * TDIM) + i];
  R[i] = s;
}

// ---- GEMM2 + correlation epilogue: out = (R @ X^T - T*mr*mx) / (nr*nx) -----
// 64(M) x 128(N loci) block tile, 16 wave32, 16x32 strip per wave.
__global__ void __launch_bounds__(512)
gemm2_corr_kernel(const float* __restrict__ X,
                  const float* __restrict__ R,
                  const float* __restrict__ mr, const float* __restrict__ nr,
                  const float* __restrict__ mx, const float* __restrict__ nx,
                  float* __restrict__ out) {
  __shared__ __align__(16) float As[64][PAD_A];    // R tile: As[m][k]
  __shared__ __align__(16) float Bs[128][PAD_B2];  // X tile: Bs[n][k]

  const int row0 = blockIdx.y * 64;     // encoding rows (pad 400 -> 448)
  const int col0 = blockIdx.x * 128;    // loci

  const int tid  = threadIdx.x;
  const int wv   = tid >> 5;
  const int lane = tid & 31;
  const int half = lane >> 4;
  const int ml   = lane & 15;
  const int mi   = wv >> 2;
  const int ni0  = (wv & 3) * 2;

  const bool rows_in = (row0 + 64 <= MTOT);
  const bool cols_in = (col0 + 128 <= N_LOCI);

  v8f acc0 = {}, acc1 = {};

  for (int kc = 0; kc < TDIM; kc += 64) {  // 8 chunks over K=500
    const bool k_in = (kc + 64 <= TDIM);
    // ---- stage A (64 x 64) ----
    if (rows_in && k_in) {
      const int m  = tid >> 3;
      const int c0 = (tid & 7) * 8;
      const float* g = R + (size_t)(row0 + m) * TDIM + kc + c0;
      ld16_to_lds(&As[m][c0], g);
      ld16_to_lds(&As[m][c0 + 4], g + 4);
    } else {
      const int m    = tid >> 3;
      const int c0   = (tid & 7) * 8;
      const int grow = row0 + m;
      const float* rr = R + (size_t)grow * TDIM;
#pragma unroll
      for (int i = 0; i < 8; ++i) {
        int k = kc + c0 + i;
        As[m][c0 + i] = (grow < MTOT && k < TDIM) ? rr[k] : 0.0f;
      }
    }
    // ---- stage B (128 loci x 64 k), contiguous-k per locus row ----
    if (cols_in && k_in) {
      const int n  = tid >> 2;
      const int c0 = (tid & 3) * 16;
      const float* g = X + (size_t)(col0 + n) * TDIM + kc + c0;
#pragma unroll
      for (int i = 0; i < 4; ++i)
        ld16_to_lds(&Bs[n][c0 + 4 * i], g + 4 * i);
    } else {
      const int n    = tid >> 2;
      const int c0   = (tid & 3) * 16;
      const int gcol = col0 + n;
      const float* xr = X + (size_t)gcol * TDIM;
#pragma unroll
      for (int i = 0; i < 16; ++i) {
        int k = kc + c0 + i;
        Bs[n][c0 + i] = (gcol < N_LOCI && k < TDIM) ? xr[k] : 0.0f;
      }
    }
    lds_copy_fence();
    __syncthreads();

#pragma unroll
    for (int k4 = 0; k4 < 64; k4 += 4) {
      const int kk = k4 + 2 * half;
      v2f a  = *(const v2f*)&As[mi * 16 + ml][kk];
      v2f b0 = *(const v2f*)&Bs[ni0 * 16 + ml][kk];        // B[k][n] = X[n][k]
      v2f b1 = *(const v2f*)&Bs[(ni0 + 1) * 16 + ml][kk];
      acc0 = __builtin_amdgcn_wmma_f32_16x16x4_f32(
          false, a, false, b0, (short)0, acc0, false, false);
      acc1 = __builtin_amdgcn_wmma_f32_16x16x4_f32(
          false, a, false, b1, (short)0, acc1, false, false);
    }
    __syncthreads();
  }

#pragma unroll
  for (int v = 0; v < 8; ++v) {
    const int r = row0 + mi * 16 + v + 8 * half;
    if (r < MTOT) {
      const float mrv = mr[r];
      const float nrv = nr[r];
      const int c0 = col0 + ni0 * 16 + ml;
      const int c1 = c0 + 16;
      if (c0 < N_LOCI) {
        float num = acc0[v] - (float)TDIM * mrv * mx[c0];
        out[(size_t)r * N_LOCI + c0] = num / (nrv * nx[c0]);
      }
      if (c1 < N_LOCI) {
        float num = acc1[v] - (float)TDIM * mrv * mx[c1];
        out[(size_t)r * N_LOCI + c1] = num / (nrv * nx[c1]);
      }
    }
  }
}

// ---------------------------------------------------------------------------
extern "C" void kernel_launch(void* const* d_in, const int* in_sizes, int n_in,
                              void* d_out, int out_size, void* d_ws,
                              size_t ws_size, hipStream_t stream) {
  const float* X  = (const float*)d_in[0];   // [59412, 500]
  const float* WL = (const float*)d_in[1];   // [200, 29696]
  const float* WR = (const float*)d_in[2];   // [200, 29716]
  float* out = (float*)d_out;                // [400, 59412]

  // workspace layout (floats): R | P[SPLITS] | mx | nx | mr | nr  (~7.7 MB)
  float* w  = (float*)d_ws;
  float* R  = w;
  float* P  = R + (size_t)MTOT * TDIM;
  float* mx = P + (size_t)SPLITS * MTOT * TDIM;
  float* nx = mx + N_LOCI;
  float* mr = nx + N_LOCI;
  float* nr = mr + MTOT;

  // 1) X row stats (mean + centered norm)
  row_stats_kernel<<<(N_LOCI + 7) / 8, 256, 0, stream>>>(X, N_LOCI, mx, nx);

  // 2) GEMM1 K-split partials: 4 col-tiles(128) x 4 row-tiles(64) x (2 comp * 8 splits)
  dim3 g1(4, 4, 2 * SPLITS);
  gemm1_partial_kernel<<<g1, 512, 0, stream>>>(X, WL, WR, P);

  // 3) deterministic reduce -> R [400, 500]
  reduce_partials_kernel<<<(MTOT * TDIM + 255) / 256, 256, 0, stream>>>(P, R);

  // 4) R row stats
  row_stats_kernel<<<(MTOT + 7) / 8, 256, 0, stream>>>(R, MTOT, mr, nr);

  // 5) big GEMM + correlation epilogue: grid = 465 x 7
  dim3 g2((N_LOCI + 127) / 128, (MTOT + 63) / 64);
  gemm2_corr_kernel<<<g2, 512, 0, stream>>>(X, R, mr, nr, mx, nx, out);
}